// HeteroSAGELayer_61435212202261
// MI455X (gfx1250) — compile-verified
//
#include <hip/hip_runtime.h>
#include <hip/hip_bf16.h>

typedef __attribute__((ext_vector_type(2))) float v2f;
typedef __attribute__((ext_vector_type(8))) float v8f;

#define NTYPES 6
#define DD     128
#define KMEAN  (NTYPES * DD)      // 768
#define KTOT   (KMEAN + DD)       // 896
#define KPAIRS (KTOT / 2)         // 448 packed k-pairs
#define ASTRIDE 900               // 900 % 64 = 4 -> rows hit distinct banks
#define HSTRIDE 132

// ---------------- kernel 1: zero the accumulators ----------------
__global__ void hsage_zero_kernel(float* __restrict__ p, size_t n) {
    size_t i = (size_t)blockIdx.x * blockDim.x + threadIdx.x;
    size_t stride = (size_t)gridDim.x * blockDim.x;
    for (; i < n; i += stride) p[i] = 0.0f;
}

// ---------------- kernel 2: build packed weights + folded bias ----------------
// Wpack[kk*128 + n] = { W(2kk, n), W(2kk+1, n) } where
//   W(k, n) = k < 768 ? W_l[k*128 + n] : sum_t W_r[t][k-768][n]
__global__ void hsage_prep_kernel(const float* __restrict__ Wl,
                                  const float* __restrict__ Wr,
                                  const float* __restrict__ b,
                                  const float* __restrict__ emb,
                                  float2* __restrict__ Wpack,
                                  float* __restrict__ bsum) {
    int i = blockIdx.x * blockDim.x + threadIdx.x;
    if (i < KPAIRS * DD) {
        int kk = i >> 7;
        int n  = i & (DD - 1);
        int k  = kk * 2;
        float v0, v1;
        if (k < KMEAN) {
            v0 = Wl[(size_t)k * DD + n];
            v1 = Wl[(size_t)(k + 1) * DD + n];
        } else {
            int d = k - KMEAN;
            v0 = 0.0f; v1 = 0.0f;
            #pragma unroll
            for (int t = 0; t < NTYPES; ++t) {
                v0 += Wr[(size_t)t * DD * DD + (size_t)d * DD + n];
                v1 += Wr[(size_t)t * DD * DD + (size_t)(d + 1) * DD + n];
            }
        }
        Wpack[i] = make_float2(v0, v1);
    }
    if (i < DD) {
        float s = 0.0f;
        #pragma unroll
        for (int t = 0; t < NTYPES; ++t) s += b[t * DD + i] + emb[t * DD + i];
        bsum[i] = s;
    }
}

// ---------------- kernel 3: edge scatter (wave32 per edge) ----------------
__global__ void hsage_scatter_kernel(const float* __restrict__ x,
                                     const int* __restrict__ edge_index,
                                     const int* __restrict__ edge_type,
                                     float* __restrict__ sums,
                                     float* __restrict__ cnt,
                                     int E) {
    int gtid = blockIdx.x * blockDim.x + threadIdx.x;
    int e    = gtid >> 5;
    int lane = gtid & 31;
    if (e >= E) return;
    int src = edge_index[e];
    int dst = edge_index[E + e];
    int t   = edge_type[e];
    int seg = dst * NTYPES + t;
    const float4 v = *(const float4*)(x + (size_t)src * DD + lane * 4);
    float* sp = sums + (size_t)seg * DD + lane * 4;
    atomicAdd(sp + 0, v.x);
    atomicAdd(sp + 1, v.y);
    atomicAdd(sp + 2, v.z);
    atomicAdd(sp + 3, v.w);
    if (lane == 0) atomicAdd(cnt + seg, 1.0f);
}

// ---------------- kernel 4: fused WMMA GEMM + LayerNorm + ReLU ----------------
// Block: 256 threads = 8 waves; computes 16 rows x 128 cols of h.
__global__ __launch_bounds__(256, 2)
void hsage_gemm_ln_kernel(const float* __restrict__ x,
                          const float* __restrict__ sums,
                          const float* __restrict__ cnt,
                          const float2* __restrict__ Wpack,  // (448, 128) k-pair packed
                          const float* __restrict__ bsum,    // (128)
                          const float* __restrict__ gamma,
                          const float* __restrict__ beta,
                          float* __restrict__ out) {
    __shared__ float Alds[16 * ASTRIDE];
    __shared__ float Hlds[16 * HSTRIDE];

    const int row0 = blockIdx.x * 16;
    const int tid  = threadIdx.x;

    // ---- stage A strip: [count-scaled means (16x768) | x (16x128)] ----
    for (int i = tid; i < 16 * KTOT; i += 256) {
        int r = i / KTOT;
        int k = i - r * KTOT;
        int gr = row0 + r;
        float v;
        if (k < KMEAN) {
            int t = k >> 7;
            int d = k & (DD - 1);
            float c   = cnt[gr * NTYPES + t];
            float inv = 1.0f / fmaxf(c, 1.0f);
            v = sums[((size_t)gr * NTYPES + t) * DD + d] * inv;
        } else {
            v = x[(size_t)gr * DD + (k - KMEAN)];
        }
        Alds[r * ASTRIDE + k] = v;
    }
    __syncthreads();

    const int wave = tid >> 5;
    const int lane = tid & 31;
    const int col0 = wave * 16;
    const int hl   = lane >> 4;   // 0: K0/K1 pair, 1: K2/K3 pair
    const int l16  = lane & 15;

    // Per-lane B pointer: fragment for step k0 is Wpack[(k0/2 + hl)*128 + n].
    const float2* wp = Wpack + (size_t)hl * DD + (col0 + l16);

    v8f c = {};
    #pragma unroll 4
    for (int k0 = 0; k0 < KTOT; k0 += 4) {
        v2f af, bf;
        const int ka = k0 + hl * 2;
        af.x = Alds[l16 * ASTRIDE + ka];
        af.y = Alds[l16 * ASTRIDE + ka + 1];
        const float2 w = wp[(size_t)(k0 >> 1) * DD];   // +2048 bytes per step
        bf.x = w.x;
        bf.y = w.y;
        c = __builtin_amdgcn_wmma_f32_16x16x4_f32(
                /*neg_a=*/false, af, /*neg_b=*/false, bf,
                /*c_mod=*/(short)0, c, /*reuse_a=*/false, /*reuse_b=*/false);
    }

    // ---- bias, spill C tile to LDS (ISA 16x16 f32 C/D layout) ----
    const float bias = bsum[col0 + l16];
    #pragma unroll
    for (int i = 0; i < 8; ++i) {
        Hlds[(i + hl * 8) * HSTRIDE + col0 + l16] = c[i] + bias;
    }
    __syncthreads();

    // ---- LayerNorm + ReLU: 16 lanes per row, 8 cols per lane ----
    const int r  = tid >> 4;
    const int cl = tid & 15;
    float vals[8];
    float s = 0.0f, ss = 0.0f;
    #pragma unroll
    for (int j = 0; j < 8; ++j) {
        float v = Hlds[r * HSTRIDE + cl * 8 + j];
        vals[j] = v;
        s  += v;
        ss += v * v;
    }
    #pragma unroll
    for (int off = 1; off < 16; off <<= 1) {   // stays within 16-lane half
        s  += __shfl_xor(s, off, 32);
        ss += __shfl_xor(ss, off, 32);
    }
    const float mu  = s * (1.0f / 128.0f);
    const float var = ss * (1.0f / 128.0f) - mu * mu;
    const float rsd = rsqrtf(var + 1e-5f);

    const size_t obase = (size_t)(row0 + r) * DD + cl * 8;
    #pragma unroll
    for (int j = 0; j < 8; ++j) {
        const int col = cl * 8 + j;
        float y = (vals[j] - mu) * rsd * gamma[col] + beta[col];
        out[obase + j] = fmaxf(y, 0.0f);
    }
}

extern "C" void kernel_launch(void* const* d_in, const int* in_sizes, int n_in,
                              void* d_out, int out_size, void* d_ws, size_t ws_size,
                              hipStream_t stream) {
    const float* x          = (const float*)d_in[0];
    const int*   edge_index = (const int*)d_in[1];
    const int*   edge_type  = (const int*)d_in[2];
    const float* W_l        = (const float*)d_in[3];
    const float* W_r        = (const float*)d_in[4];
    const float* b          = (const float*)d_in[5];
    const float* emb        = (const float*)d_in[6];
    const float* gamma      = (const float*)d_in[7];
    const float* beta       = (const float*)d_in[8];
    float*       out        = (float*)d_out;

    const int N = in_sizes[0] / DD;       // 50000
    const int E = in_sizes[2];            // 600000

    // workspace layout (floats)
    float*  ws    = (float*)d_ws;
    float*  sums  = ws;                                   // N*T*128
    float*  cnt   = sums + (size_t)N * NTYPES * DD;       // N*T
    float2* Wpack = (float2*)(cnt + (size_t)N * NTYPES);  // 448*128 float2
    float*  bsum  = (float*)(Wpack + (size_t)KPAIRS * DD);// 128

    const size_t nzero = (size_t)N * NTYPES * DD + (size_t)N * NTYPES;
    hsage_zero_kernel<<<2048, 256, 0, stream>>>(sums, nzero);

    hsage_prep_kernel<<<(KPAIRS * DD + 255) / 256, 256, 0, stream>>>(
        W_l, W_r, b, emb, Wpack, bsum);

    const int scatter_threads = E * 32;
    hsage_scatter_kernel<<<(scatter_threads + 255) / 256, 256, 0, stream>>>(
        x, edge_index, edge_type, sums, cnt, E);

    hsage_gemm_ln_kernel<<<N / 16, 256, 0, stream>>>(
        x, sums, cnt, Wpack, bsum, gamma, beta, out);
}